// BlockConv_47459388620823
// MI455X (gfx1250) — compile-verified
//
#include <hip/hip_runtime.h>
#include <cstdint>

#define NN 50000
#define NE 800000
#define C 64
#define EPS 1e-5f

typedef __attribute__((ext_vector_type(2))) float v2f;
typedef __attribute__((ext_vector_type(8))) float v8f;

// Float atomic max via sign-split integer atomics (valid with -inf init):
// positive floats order like ints, negative floats order inversely like uints.
__device__ __forceinline__ void atomic_max_f32(float* addr, float val) {
  if (val >= 0.0f)
    atomicMax((int*)addr, __float_as_int(val));
  else
    atomicMin((unsigned int*)addr, __float_as_uint(val));
}

// ---- zero the 3 x (sum,sumsq) BN accumulators (384 floats) ----
__global__ void zero_stats_kernel(float* __restrict__ stats) {
  int i = threadIdx.x;
  if (i < 384) stats[i] = 0.0f;
}

// ---- fill segment-max buffer with -inf ----
__global__ void init_neg_inf_kernel(float* __restrict__ tmp, int n) {
  int i = blockIdx.x * blockDim.x + threadIdx.x;
  if (i < n) tmp[i] = -__builtin_inff();
}

// ---- node-level GEMM via V_WMMA_F32_16X16X4_F32 ----
// out[n][c] = sum_k in[n][k]*W[k][c]  (+ pos2[n]@W[64:66] if pos)  (+ bias[c] if bias)
// Block = 128 threads = 4 waves; wave w owns the 16x16 tile at cols w*16.
// Grid.x = NN/16 (NN divisible by 16 -> EXEC all ones, as WMMA requires).
__global__ void node_gemm_kernel(const float* __restrict__ in,
                                 const float* __restrict__ W,     // row-major, ld = 64
                                 const float* __restrict__ pos,   // (NN,3) or nullptr
                                 const float* __restrict__ bias,  // (64,) or nullptr
                                 float* __restrict__ out) {
  const int wave = threadIdx.x >> 5;
  const int lane = threadIdx.x & 31;
  const int lo = lane & 15;   // M index (A) / N index (B,C,D)
  const int hi = lane >> 4;   // K half-select (A,B) / M+8 select (C,D)
  const int m0 = blockIdx.x * 16;
  const int n0 = wave * 16;

  v8f acc = {};
#pragma unroll
  for (int k0 = 0; k0 < C; k0 += 4) {
    const int ka = k0 + hi * 2;  // VGPR0 holds K = {0,2}+k0, VGPR1 holds K = {1,3}+k0
    v2f a, b;
    a.x = in[(m0 + lo) * C + ka];
    a.y = in[(m0 + lo) * C + ka + 1];
    b.x = W[ka * C + (n0 + lo)];
    b.y = W[(ka + 1) * C + (n0 + lo)];
    acc = __builtin_amdgcn_wmma_f32_16x16x4_f32(
        /*neg_a=*/false, a, /*neg_b=*/false, b,
        /*c_mod=*/(short)0, acc, /*reuse_a=*/false, /*reuse_b=*/false);
  }

#pragma unroll
  for (int r = 0; r < 8; ++r) {
    const int row = m0 + hi * 8 + r;   // C/D layout: VGPR r -> M=r (lanes 0-15), M=r+8 (lanes 16-31)
    const int col = n0 + lo;
    float v = acc[r];
    if (pos)  // pos2 @ W[64:66]
      v += pos[row * 3 + 0] * W[64 * C + col] + pos[row * 3 + 1] * W[65 * C + col];
    if (bias) v += bias[col];
    out[row * C + col] = v;
  }
}

// ---- per-edge segment max: tmp[dst][:] = max(tmp[dst][:], s[src][:]) ----
// 16 lanes per edge, 4 channels each (coalesced 256B row gather -> hits L2).
__global__ void edge_scatter_max_kernel(const long long* __restrict__ ei,  // int64 (2,E)
                                        const float* __restrict__ s,
                                        float* __restrict__ tmp) {
  const long long t = (long long)blockIdx.x * blockDim.x + threadIdx.x;
  const int e = (int)(t >> 4);
  const int l = (int)(t & 15);
  if (e >= NE) return;
  const int src = (int)ei[e];
  const int dst = (int)ei[NE + e];
  const float4 v = *(const float4*)(s + src * C + l * 4);
  float* base = tmp + dst * C + l * 4;
  atomic_max_f32(base + 0, v.x);
  atomic_max_f32(base + 1, v.y);
  atomic_max_f32(base + 2, v.z);
  atomic_max_f32(base + 3, v.w);
}

// ---- fold per-dst constant back in: h = (tmp==-inf) ? 0 : tmp - pos2@Wp + b ----
__global__ void combine_conv_kernel(const float* __restrict__ tmp,
                                    const float* __restrict__ W,   // (66,64)
                                    const float* __restrict__ bias,
                                    const float* __restrict__ pos,
                                    float* __restrict__ h) {
  const int i = blockIdx.x * blockDim.x + threadIdx.x;
  if (i >= NN * C) return;
  const int row = i >> 6, c = i & 63;
  const float v = tmp[i];
  float o;
  if (v == -__builtin_inff()) {
    o = 0.0f;  // isolated node: reference maps -inf -> 0
  } else {
    o = v - (pos[row * 3 + 0] * W[64 * C + c] + pos[row * 3 + 1] * W[65 * C + c]) + bias[c];
  }
  h[i] = o;
}

// ---- per-channel sum / sumsq reduction (128 rows per block, LDS then atomics) ----
__global__ void reduce_sums_kernel(const float* __restrict__ m,
                                   float* __restrict__ stats) {  // [0:64]=sum [64:128]=sumsq
  __shared__ float ls[256];
  __shared__ float lq[256];
  const int c = threadIdx.x & 63;
  const int g = threadIdx.x >> 6;
  const int base = blockIdx.x * 128;
  float s = 0.0f, q = 0.0f;
  for (int r = base + g; r < base + 128 && r < NN; r += 4) {
    const float v = m[r * C + c];
    s += v;
    q += v * v;
  }
  ls[threadIdx.x] = s;
  lq[threadIdx.x] = q;
  __syncthreads();
  if (g == 0) {
    s = ls[c] + ls[64 + c] + ls[128 + c] + ls[192 + c];
    q = lq[c] + lq[64 + c] + lq[128 + c] + lq[192 + c];
    atomicAdd(&stats[c], s);
    atomicAdd(&stats[64 + c], q);
  }
}

// ---- h = relu(BN(h)) in-place ----
__global__ void bn_relu_kernel(float* __restrict__ h, const float* __restrict__ stats,
                               const float* __restrict__ gamma,
                               const float* __restrict__ beta) {
  const int i = blockIdx.x * blockDim.x + threadIdx.x;
  if (i >= NN * C) return;
  const int c = i & 63;
  const float inv_n = 1.0f / (float)NN;
  const float mean = stats[c] * inv_n;
  const float var = stats[64 + c] * inv_n - mean * mean;
  const float rs = rsqrtf(var + EPS);
  const float v = gamma[c] * (h[i] - mean) * rs + beta[c];
  h[i] = fmaxf(v, 0.0f);
}

// ---- out = relu(BN2(h2) + BNl(skip)) ----
__global__ void final_kernel(const float* __restrict__ h2, const float* __restrict__ st2,
                             const float* __restrict__ g2, const float* __restrict__ be2,
                             const float* __restrict__ skip, const float* __restrict__ stL,
                             const float* __restrict__ gl, const float* __restrict__ bel,
                             float* __restrict__ out) {
  const int i = blockIdx.x * blockDim.x + threadIdx.x;
  if (i >= NN * C) return;
  const int c = i & 63;
  const float inv_n = 1.0f / (float)NN;
  const float m2 = st2[c] * inv_n;
  const float v2 = st2[64 + c] * inv_n - m2 * m2;
  const float a = g2[c] * (h2[i] - m2) * rsqrtf(v2 + EPS) + be2[c];
  const float mL = stL[c] * inv_n;
  const float vL = stL[64 + c] * inv_n - mL * mL;
  const float b = gl[c] * (skip[i] - mL) * rsqrtf(vL + EPS) + bel[c];
  out[i] = fmaxf(a + b, 0.0f);
}

extern "C" void kernel_launch(void* const* d_in, const int* in_sizes, int n_in,
                              void* d_out, int out_size, void* d_ws, size_t ws_size,
                              hipStream_t stream) {
  (void)in_sizes; (void)n_in; (void)out_size; (void)ws_size;
  const float* x       = (const float*)d_in[0];
  const float* pos     = (const float*)d_in[1];
  const long long* ei  = (const long long*)d_in[2];  // int64 edge_index
  const float* w1      = (const float*)d_in[3];
  const float* b1      = (const float*)d_in[4];
  const float* g1      = (const float*)d_in[5];
  const float* be1     = (const float*)d_in[6];
  const float* w2      = (const float*)d_in[7];
  const float* b2      = (const float*)d_in[8];
  const float* g2      = (const float*)d_in[9];
  const float* be2     = (const float*)d_in[10];
  const float* wl      = (const float*)d_in[11];
  const float* bl      = (const float*)d_in[12];
  const float* gl      = (const float*)d_in[13];
  const float* bel     = (const float*)d_in[14];
  float* out = (float*)d_out;

  const size_t NC = (size_t)NN * C;
  float* skip  = (float*)d_ws;     // BN-input of linear skip path
  float* sbuf  = skip + NC;        // per-node conv messages s = x@Wa + pos2@Wp
  float* tmpb  = sbuf + NC;        // segment-max accumulator
  float* hbuf  = tmpb + NC;        // conv output / hidden
  float* statL = hbuf + NC;        // 128 floats
  float* stat1 = statL + 128;      // 128 floats
  float* stat2 = stat1 + 128;      // 128 floats

  const dim3 gGemm(NN / 16), bGemm(128);
  const int nElem = NN * C;
  const int gElem = (nElem + 255) / 256;
  const int gRed = (NN + 127) / 128;
  const long long scT = (long long)NE * 16;
  const int gScatter = (int)((scT + 255) / 256);

  zero_stats_kernel<<<1, 384, 0, stream>>>(statL);

  // skip path: skip = x @ wl + bl, then channel stats
  node_gemm_kernel<<<gGemm, bGemm, 0, stream>>>(x, wl, nullptr, bl, skip);
  reduce_sums_kernel<<<gRed, 256, 0, stream>>>(skip, statL);

  // conv layer 1
  node_gemm_kernel<<<gGemm, bGemm, 0, stream>>>(x, w1, pos, nullptr, sbuf);
  init_neg_inf_kernel<<<gElem, 256, 0, stream>>>(tmpb, nElem);
  edge_scatter_max_kernel<<<gScatter, 256, 0, stream>>>(ei, sbuf, tmpb);
  combine_conv_kernel<<<gElem, 256, 0, stream>>>(tmpb, w1, b1, pos, hbuf);
  reduce_sums_kernel<<<gRed, 256, 0, stream>>>(hbuf, stat1);
  bn_relu_kernel<<<gElem, 256, 0, stream>>>(hbuf, stat1, g1, be1);

  // conv layer 2
  node_gemm_kernel<<<gGemm, bGemm, 0, stream>>>(hbuf, w2, pos, nullptr, sbuf);
  init_neg_inf_kernel<<<gElem, 256, 0, stream>>>(tmpb, nElem);
  edge_scatter_max_kernel<<<gScatter, 256, 0, stream>>>(ei, sbuf, tmpb);
  combine_conv_kernel<<<gElem, 256, 0, stream>>>(tmpb, w2, b2, pos, hbuf);
  reduce_sums_kernel<<<gRed, 256, 0, stream>>>(hbuf, stat2);

  // out = relu(BN2(h2) + BNl(skip))
  final_kernel<<<gElem, 256, 0, stream>>>(hbuf, stat2, g2, be2, skip, statL, gl, bel, out);
}